// A100OptimizedSparseSimilarity_80135499809328
// MI455X (gfx1250) — compile-verified
//
#include <hip/hip_runtime.h>
#include <hip/hip_bf16.h>
#include <math.h>

// ---------------------------------------------------------------------------
// sim = normalize(x) @ normalize(y)^T / tau ; per-row top-10 ; softmax ; COO
//   x,y : [16384, 512] f32.  Fused WMMA-f32 GEMM + streaming exact top-k.
// ---------------------------------------------------------------------------

#define NROWS 16384
#define NCOLSY 16384
#define CDIM 512
#define TOPK 10
#define TAU_INV 20.0f        // 1 / 0.05
#define BLK_ROWS 16          // x-rows per workgroup
#define NWAVES 8             // 256 threads, wave32
#define CHUNK 128            // y-cols per chunk = NWAVES * 16
#define SIM_PITCH 132        // padded LDS pitch (floats) to dodge bank conflicts

typedef __attribute__((ext_vector_type(2))) float v2f;
typedef __attribute__((ext_vector_type(8))) float v8f;

// ---- Pass 1: inverse L2 norms of every row of x and y -> workspace --------
__global__ __launch_bounds__(128) void inv_norm_kernel(
    const float* __restrict__ x, const float* __restrict__ y,
    float* __restrict__ invx, float* __restrict__ invy) {
  const int r = blockIdx.x;  // 0..2*NROWS-1
  const float* src =
      (r < NROWS) ? (x + (size_t)r * CDIM) : (y + (size_t)(r - NROWS) * CDIM);
  const int tid = threadIdx.x;
  float4 v = ((const float4*)src)[tid];  // 128 threads * 4 floats = 512
  float s = v.x * v.x + v.y * v.y + v.z * v.z + v.w * v.w;
#pragma unroll
  for (int off = 16; off > 0; off >>= 1) s += __shfl_xor(s, off, 32);
  __shared__ float wsum[4];
  if ((tid & 31) == 0) wsum[tid >> 5] = s;
  __syncthreads();
  if (tid == 0) {
    float t = wsum[0] + wsum[1] + wsum[2] + wsum[3];
    float inv = 1.0f / fmaxf(sqrtf(t), 1e-12f);
    if (r < NROWS) invx[r] = inv; else invy[r - NROWS] = inv;
  }
}

// ---- Pass 2: fused GEMM (v_wmma_f32_16x16x4_f32) + streaming top-10 -------
__global__ __launch_bounds__(256) void simtopk_kernel(
    const float* __restrict__ x, const float* __restrict__ y,
    const float* __restrict__ invx, const float* __restrict__ invy,
    float* __restrict__ out_vals, int* __restrict__ out_rows,
    int* __restrict__ out_cols) {
  __shared__ float lds_a[BLK_ROWS * CDIM];        // 32 KB  A tile (raw x)
  __shared__ float sim[BLK_ROWS * SIM_PITCH];     // 8.25 KB sim strip
  __shared__ float topv[BLK_ROWS][TOPK];
  __shared__ int   topi[BLK_ROWS][TOPK];

  const int tid = threadIdx.x;
  const int wave = tid >> 5;
  const int lane = tid & 31;
  const int rowbase = blockIdx.x * BLK_ROWS;

  // Stage the 16x512 A tile into LDS (coalesced b128).
  {
    const float4* src = (const float4*)(x + (size_t)rowbase * CDIM);
    float4* dst = (float4*)lds_a;
    for (int i = tid; i < BLK_ROWS * CDIM / 4; i += 256) dst[i] = src[i];
  }
  for (int i = tid; i < BLK_ROWS * TOPK; i += 256) {
    ((float*)topv)[i] = -INFINITY;
    ((int*)topi)[i] = 0;
  }
  __syncthreads();

  // WMMA 16x16x4 f32 operand layout (ISA 7.12.2):
  //   A 16x4: lanes 0-15 -> M=lane, VGPR0/1 = K0/K1; lanes 16-31 -> K2/K3.
  //   B 4x16: same K-pair split (rows v, v+... style), N = lane&15.
  //   C/D:    VGPR i -> M = i + 8*(lane>>4), N = lane&15.
  const int n = lane & 15;
  const int h = lane >> 4;
  const int koff = h * 2;
  const float* aptr = lds_a + n * CDIM + koff;  // A row M = lane&15

  float invm[8];  // C-scale: invx for the 8 M-rows this lane accumulates
#pragma unroll
  for (int i = 0; i < 8; ++i) invm[i] = invx[rowbase + h * 8 + i];

  for (int c = 0; c < NCOLSY / CHUNK; ++c) {
    const int colbase = c * CHUNK;
    const int col = colbase + wave * 16 + n;
    const float* bptr = y + (size_t)col * CDIM + koff;

    v8f acc0 = {};
    v8f acc1 = {};
#pragma unroll 4
    for (int kb = 0; kb < CDIM; kb += 8) {
      v2f a0 = *(const v2f*)(aptr + kb);
      v2f b0 = *(const v2f*)(bptr + kb);
      acc0 = __builtin_amdgcn_wmma_f32_16x16x4_f32(
          false, a0, false, b0, (short)0, acc0, false, false);
      v2f a1 = *(const v2f*)(aptr + kb + 4);
      v2f b1 = *(const v2f*)(bptr + kb + 4);
      acc1 = __builtin_amdgcn_wmma_f32_16x16x4_f32(
          false, a1, false, b1, (short)0, acc1, false, false);
    }

    const float sy = invy[col];
#pragma unroll
    for (int i = 0; i < 8; ++i) {
      float v = (acc0[i] + acc1[i]) * (invm[i] * sy);
      sim[(h * 8 + i) * SIM_PITCH + wave * 16 + n] = v;
    }
    __syncthreads();

    // Each wave maintains exact top-10 for its two rows.
    for (int rr = 0; rr < 2; ++rr) {
      const int r = wave * 2 + rr;
      float lv[4];
#pragma unroll
      for (int j = 0; j < 4; ++j) lv[j] = sim[r * SIM_PITCH + j * 32 + lane];

      for (int it = 0; it < TOPK; ++it) {
        float mv = lv[0];
        int mj = 0;
#pragma unroll
        for (int j = 1; j < 4; ++j)
          if (lv[j] > mv) { mv = lv[j]; mj = j; }
        int mi = mj * 32 + lane;
#pragma unroll
        for (int off = 16; off > 0; off >>= 1) {  // argmax, lower idx on ties
          float ov = __shfl_xor(mv, off, 32);
          int oi = __shfl_xor(mi, off, 32);
          if (ov > mv || (ov == mv && oi < mi)) { mv = ov; mi = oi; }
        }
        if (mv <= topv[r][TOPK - 1]) break;  // cannot enter the list
        if (lane == 0) {                      // sorted insert (desc)
          int p = TOPK - 1;
          while (p > 0 && topv[r][p - 1] < mv) {
            topv[r][p] = topv[r][p - 1];
            topi[r][p] = topi[r][p - 1];
            --p;
          }
          topv[r][p] = mv;
          topi[r][p] = colbase + mi;
        }
        if ((mi & 31) == lane) lv[mi >> 5] = -INFINITY;  // remove winner
      }
    }
    __syncthreads();
  }

  // Softmax over the 10 survivors (tau applied here; monotone so top-k safe).
  for (int rr = 0; rr < 2; ++rr) {
    const int r = wave * 2 + rr;
    const int R = rowbase + r;
    const float vmax = topv[r][0];  // sorted descending
    float e = 0.0f;
    if (lane < TOPK) e = __expf((topv[r][lane] - vmax) * TAU_INV);
    float s = e;
#pragma unroll
    for (int off = 16; off > 0; off >>= 1) s += __shfl_xor(s, off, 32);
    if (lane < TOPK) {
      out_vals[R * TOPK + lane] = e / s;
      out_rows[R * TOPK + lane] = R;            // indices[0, :]
      out_cols[R * TOPK + lane] = topi[r][lane];  // indices[1, :]
    }
  }
}

extern "C" void kernel_launch(void* const* d_in, const int* in_sizes, int n_in,
                              void* d_out, int out_size, void* d_ws,
                              size_t ws_size, hipStream_t stream) {
  (void)in_sizes; (void)n_in; (void)out_size; (void)ws_size;
  const float* fx = (const float*)d_in[0];
  const float* fy = (const float*)d_in[1];

  float* invx = (float*)d_ws;            // 16384 floats
  float* invy = invx + NROWS;            // 16384 floats  (128 KB total)

  inv_norm_kernel<<<2 * NROWS, 128, 0, stream>>>(fx, fy, invx, invy);

  // Output: values f32 [163840], then indices i32 [2,163840] flat.
  float* out_vals = (float*)d_out;
  int* out_idx = (int*)((float*)d_out + (size_t)NROWS * TOPK);
  int* out_rows = out_idx;
  int* out_cols = out_idx + (size_t)NROWS * TOPK;

  simtopk_kernel<<<NROWS / BLK_ROWS, 256, 0, stream>>>(
      fx, fy, invx, invy, out_vals, out_rows, out_cols);
}